// AveragedHausdorffLoss_76716705841675
// MI455X (gfx1250) — compile-verified
//
#include <hip/hip_runtime.h>
#include <hip/hip_bf16.h>
#include <math.h>

typedef __attribute__((ext_vector_type(2))) float v2f;
typedef __attribute__((ext_vector_type(8))) float v8f;

// Prepass: pack point i as (x, y, z, |p|^2).
// Memory layout per point = two float2 halves: (x,y) and (z,|p|^2) — exactly the
// two B-operand register values needed by the lo/hi lane halves of the WMMA.
__global__ __launch_bounds__(256) void pack_kernel(
    const float* __restrict__ P, float4* __restrict__ out, int n)
{
    int i = blockIdx.x * blockDim.x + threadIdx.x;
    if (i < n) {
        float x = P[3 * i], y = P[3 * i + 1], z = P[3 * i + 2];
        out[i] = make_float4(x, y, z, x * x + y * y + z * z);
    }
}

// One wave owns a 16-row tile of Q and a segment of T tiles, computing
// rm[seg][m] = min over its T segment of ||Q_m - T_j||^2 via
// V_WMMA_F32_16X16X4_F32 with the squared-norm fold in K=3:
//   A[m][0..2] = -2*q_m,  A[m][3] = 1
//   B[0..2][n] =  t_n,    B[3][n] = |t_n|^2
//   C[m][n]    = |q_m|^2
// => D[m][n] = |q|^2 - 2 q.t + |t|^2 = d^2 directly; epilogue is pure min.
__global__ __launch_bounds__(256) void hausdorff_rowmin_kernel(
    const float4* __restrict__ Q4, const float4* __restrict__ T4,
    float* __restrict__ rm_out, int nq, int nt, int split)
{
    const int wavesPerBlock = blockDim.x >> 5;
    const int mtilesTotal   = nq >> 4;
    const int bpp  = (mtilesTotal + wavesPerBlock - 1) / wavesPerBlock; // blocks per segment
    const int seg  = blockIdx.x / bpp;
    const int blk  = blockIdx.x % bpp;

    const int lane  = threadIdx.x & 31;
    const int wave  = threadIdx.x >> 5;
    const int mt    = blk * wavesPerBlock + wave;     // wave-uniform
    const int mbase = mt * 16;
    if (mbase >= nq) return;                          // wave-uniform: EXEC stays all-1s

    const int  lmod = lane & 15;
    const bool hi   = lane >= 16;

    // Q tile point for row mbase+lmod (both halves load the same row).
    const float4 q4 = Q4[mbase + lmod];

    // A tile. Lane l, VGPR v: K = v + 2*hi.  K=3 slot holds 1.0.
    v2f a;
    a.x = -2.0f * (hi ? q4.z : q4.x);
    a.y = hi ? 1.0f : (-2.0f * q4.y);

    // C init: cinit[v] = |Q_{mbase + v + 8*hi}|^2  (C/D layout: M = v + 8*(lane>=16))
    v8f cinit;
    #pragma unroll
    for (int v = 0; v < 8; ++v)
        cinit[v] = __shfl(q4.w, v + (hi ? 8 : 0), 32);

    v8f rm;
    #pragma unroll
    for (int v = 0; v < 8; ++v) rm[v] = 3.0e38f;

    // This wave's segment of T tiles.
    const int ntiles = nt >> 4;
    const int per    = ntiles / split;
    const int t0     = seg * per;
    const int t1     = (seg == split - 1) ? ntiles : (t0 + per);

    // Each lane loads only the float2 half it feeds into B: lo half -> (x,y),
    // hi half -> (z,|t|^2).  One global_load_b64 per lane per tile, no cndmask.
    const float2* tp = (const float2*)T4 + (((size_t)t0 * 16 + lmod) << 1) + (hi ? 1 : 0);

    #pragma unroll 4
    for (int t = t0; t < t1; ++t, tp += 32) {
        const float2 b2 = *tp;                // global_load_b64, cache-resident
        v2f b;
        b.x = b2.x;
        b.y = b2.y;

        // 8 args: (neg_a, A, neg_b, B, c_mod, C, reuse_a, reuse_b)
        v8f c = __builtin_amdgcn_wmma_f32_16x16x4_f32(
            false, a, false, b, (short)0, cinit, false, false);

        #pragma unroll
        for (int v = 0; v < 8; ++v)
            rm[v] = fminf(rm[v], c[v]);       // c[v] is already squared distance
    }

    // Min-reduce each rm[v] across its 16-lane half; lanes 0/16 write rows v / v+8.
    #pragma unroll
    for (int v = 0; v < 8; ++v) {
        float r = rm[v];
        r = fminf(r, __shfl_xor(r, 8, 32));
        r = fminf(r, __shfl_xor(r, 4, 32));
        r = fminf(r, __shfl_xor(r, 2, 32));
        r = fminf(r, __shfl_xor(r, 1, 32));
        if (lmod == 0)
            rm_out[(size_t)seg * nq + mbase + v + (hi ? 8 : 0)] = r;
    }
}

// Stage 1: each block deterministically sums sqrt(min_seg rm) over its strided
// share of both arrays, writes one partial per block.
__global__ __launch_bounds__(256) void hausdorff_finalize1_kernel(
    const float* __restrict__ rm1, const float* __restrict__ rm2,
    float* __restrict__ partials, int n1, int n2, int split)
{
    __shared__ float sdata[256];
    const float inv1 = 1.0f / (float)n1;
    const float inv2 = 1.0f / (float)n2;
    const int stride = gridDim.x * 256;
    float s = 0.0f;
    for (int i = blockIdx.x * 256 + threadIdx.x; i < n1; i += stride) {
        float r = 3.0e38f;
        for (int g = 0; g < split; ++g) r = fminf(r, rm1[(size_t)g * n1 + i]);
        s += sqrtf(fmaxf(r, 0.0f)) * inv1;
    }
    for (int i = blockIdx.x * 256 + threadIdx.x; i < n2; i += stride) {
        float r = 3.0e38f;
        for (int g = 0; g < split; ++g) r = fminf(r, rm2[(size_t)g * n2 + i]);
        s += sqrtf(fmaxf(r, 0.0f)) * inv2;
    }
    sdata[threadIdx.x] = s;
    __syncthreads();
    for (int off = 128; off > 0; off >>= 1) {
        if ((int)threadIdx.x < off) sdata[threadIdx.x] += sdata[threadIdx.x + off];
        __syncthreads();
    }
    if (threadIdx.x == 0) partials[blockIdx.x] = sdata[0];
}

// Stage 2: single block sums the per-block partials.
__global__ __launch_bounds__(256) void hausdorff_finalize2_kernel(
    const float* __restrict__ partials, float* __restrict__ out, int nb)
{
    __shared__ float sdata[256];
    float s = ((int)threadIdx.x < nb) ? partials[threadIdx.x] : 0.0f;
    sdata[threadIdx.x] = s;
    __syncthreads();
    for (int off = 128; off > 0; off >>= 1) {
        if ((int)threadIdx.x < off) sdata[threadIdx.x] += sdata[threadIdx.x + off];
        __syncthreads();
    }
    if (threadIdx.x == 0) out[0] = sdata[0];
}

extern "C" void kernel_launch(void* const* d_in, const int* in_sizes, int n_in,
                              void* d_out, int out_size, void* d_ws, size_t ws_size,
                              hipStream_t stream) {
    const float* set1 = (const float*)d_in[0];
    const float* set2 = (const float*)d_in[1];
    const int n1 = in_sizes[0] / 3;   // 16384
    const int n2 = in_sizes[1] / 3;   // 16384
    const int NB_FINAL = 64;          // finalize stage-1 blocks

    // Pick the largest T-segment split whose partial-min arrays fit in d_ws.
    const size_t packBytes = (size_t)(n1 + n2) * sizeof(float4);
    int split = 1;
    for (int s = 8; s >= 1; s >>= 1) {
        size_t need = packBytes + (size_t)s * (n1 + n2) * sizeof(float)
                    + NB_FINAL * sizeof(float);
        if (need <= ws_size) { split = s; break; }
    }

    char* ws = (char*)d_ws;
    float4* Q4p = (float4*)ws;                                    // n1 float4
    float4* T4p = Q4p + n1;                                       // n2 float4
    float*  rm1 = (float*)(T4p + n2);                             // split*n1 floats
    float*  rm2 = rm1 + (size_t)split * n1;                       // split*n2 floats
    float*  partials = rm2 + (size_t)split * n2;                  // NB_FINAL floats

    const int threads = 256;                  // 8 waves per workgroup
    const int wavesPerBlock = threads / 32;

    pack_kernel<<<(n1 + threads - 1) / threads, threads, 0, stream>>>(set1, Q4p, n1);
    pack_kernel<<<(n2 + threads - 1) / threads, threads, 0, stream>>>(set2, T4p, n2);

    const int bpp1 = (n1 / 16 + wavesPerBlock - 1) / wavesPerBlock;   // 128
    const int bpp2 = (n2 / 16 + wavesPerBlock - 1) / wavesPerBlock;   // 128

    hausdorff_rowmin_kernel<<<bpp1 * split, threads, 0, stream>>>(Q4p, T4p, rm1, n1, n2, split);
    hausdorff_rowmin_kernel<<<bpp2 * split, threads, 0, stream>>>(T4p, Q4p, rm2, n2, n1, split);
    hausdorff_finalize1_kernel<<<NB_FINAL, threads, 0, stream>>>(rm1, rm2, partials, n1, n2, split);
    hausdorff_finalize2_kernel<<<1, threads, 0, stream>>>(partials, (float*)d_out, NB_FINAL);
}